// SphericalHarmonics_75866302317022
// MI455X (gfx1250) — compile-verified
//
#include <hip/hip_runtime.h>

typedef __attribute__((ext_vector_type(2))) float v2f;
typedef __attribute__((ext_vector_type(8))) float v8f;

// Constant coefficient matrix M[16 outputs][16 features].
// Features per point:
//  f0=1 f1=ct f2=ct^2 f3=ct^3
//  f4=cp*st f5=cp*st*ct f6=cp*st*ct^2
//  f7=sp*st f8=sp*st*ct f9=sp*st*ct^2
//  f10=c2p*st^2 f11=c2p*st^2*ct
//  f12=s2p*st^2 f13=s2p*st^2*ct
//  f14=c3p*st^3 f15=s3p*st^3
// Outputs (rows): Y0[0]; Y1[-1,0,1]; Y2[-2..2]; Y3[-3..3]  (reference sign convention)
__constant__ float Mmat[16][16] = {
  /* Y0  m=0  */ {0.28209479177387814f,0,0,0, 0,0,0, 0,0,0, 0,0, 0,0, 0,0},
  /* Y1 m=-1 */ {0,0,0,0, 0,0,0, -0.4886025119029199f,0,0, 0,0, 0,0, 0,0},
  /* Y1 m=0  */ {0,0.4886025119029199f,0,0, 0,0,0, 0,0,0, 0,0, 0,0, 0,0},
  /* Y1 m=+1 */ {0,0,0,0, -0.4886025119029199f,0,0, 0,0,0, 0,0, 0,0, 0,0},
  /* Y2 m=-2 */ {0,0,0,0, 0,0,0, 0,0,0, 0,0, 0.5462742152960396f,0, 0,0},
  /* Y2 m=-1 */ {0,0,0,0, 0,0,0, 0,-1.0925484305920792f,0, 0,0, 0,0, 0,0},
  /* Y2 m=0  */ {-0.31539156525252005f,0,0.9461746957575601f,0, 0,0,0, 0,0,0, 0,0, 0,0, 0,0},
  /* Y2 m=+1 */ {0,0,0,0, 0,-1.0925484305920792f,0, 0,0,0, 0,0, 0,0, 0,0},
  /* Y2 m=+2 */ {0,0,0,0, 0,0,0, 0,0,0, 0.5462742152960396f,0, 0,0, 0,0},
  /* Y3 m=-3 */ {0,0,0,0, 0,0,0, 0,0,0, 0,0, 0,0, 0,-0.5900435899266435f},
  /* Y3 m=-2 */ {0,0,0,0, 0,0,0, 0,0,0, 0,0, 0,1.445305721320277f, 0,0},
  /* Y3 m=-1 */ {0,0,0,0, 0,0,0, 0.4570457994644658f,0,-2.285228997322329f, 0,0, 0,0, 0,0},
  /* Y3 m=0  */ {0,-1.1195289977703462f,0,1.865881662950577f, 0,0,0, 0,0,0, 0,0, 0,0, 0,0},
  /* Y3 m=+1 */ {0,0,0,0, 0.4570457994644658f,0,-2.285228997322329f, 0,0,0, 0,0, 0,0, 0,0},
  /* Y3 m=+2 */ {0,0,0,0, 0,0,0, 0,0,0, 0,1.445305721320277f, 0,0, 0,0},
  /* Y3 m=+3 */ {0,0,0,0, 0,0,0, 0,0,0, 0,0, 0,0, -0.5900435899266435f,0},
};

__global__ __launch_bounds__(256) void sph_wmma_kernel(const float* __restrict__ d,
                                                       float* __restrict__ out,
                                                       int N) {
    const int lane = threadIdx.x & 31;
    const int wid  = (int)((blockIdx.x * blockDim.x + threadIdx.x) >> 5);
    const long long base = (long long)wid * 32;

    // ---- per-lane feature computation (one point per lane) ----
    long long p  = base + lane;
    long long pl = (p < N) ? p : (long long)N - 1;   // clamp OOB loads
    const float phi = __builtin_nontemporal_load(d + 3 * pl + 1);
    const float th  = __builtin_nontemporal_load(d + 3 * pl + 2);

    const float ct = cosf(th);
    const float st = sqrtf(1.0f - ct * ct);          // matches reference
    const float cp = cosf(phi);
    const float sp = sinf(phi);
    const float c2 = cp * cp - sp * sp;
    const float s2 = 2.0f * sp * cp;
    const float c3 = c2 * cp - s2 * sp;
    const float s3 = s2 * cp + c2 * sp;
    const float ct2 = ct * ct, ct3 = ct2 * ct;
    const float st2 = st * st, st3 = st2 * st;

    float fr[16];
    fr[0] = 1.0f;     fr[1] = ct;        fr[2] = ct2;       fr[3] = ct3;
    fr[4] = cp * st;  fr[5] = fr[4]*ct;  fr[6] = fr[4]*ct2;
    fr[7] = sp * st;  fr[8] = fr[7]*ct;  fr[9] = fr[7]*ct2;
    fr[10] = c2*st2;  fr[11] = fr[10]*ct;
    fr[12] = s2*st2;  fr[13] = fr[12]*ct;
    fr[14] = c3*st3;  fr[15] = s3*st3;

    // ---- A operand: constant matrix in WMMA 16x4-chunk layout ----
    // A chunk c: lanes 0-15 hold row=lane, (K=4c+0, 4c+1); lanes 16-31 hold (K=4c+2, 4c+3)
    const int row = lane & 15;
    const int hi2 = (lane >> 4) << 1;
    v2f A[4];
#pragma unroll
    for (int c = 0; c < 4; ++c) {
        A[c].x = Mmat[row][4 * c + hi2 + 0];
        A[c].y = Mmat[row][4 * c + hi2 + 1];
    }

    const bool hi = (lane >= 16);
    const size_t Nn = (size_t)N;
    float* __restrict__ out1 = out + Nn;        // Y1: (N,3)
    float* __restrict__ out2 = out + 4 * Nn;    // Y2: (N,5)
    float* __restrict__ out3 = out + 9 * Nn;    // Y3: (N,7)

    // ---- two 16-point groups per wave ----
#pragma unroll
    for (int g = 0; g < 2; ++g) {
        const int src = (g << 4) | row;   // lane holding point for column n = row
        v8f acc = {};
#pragma unroll
        for (int c = 0; c < 4; ++c) {
            // B chunk c (4x16): lanes 0-15 carry (K=4c+0, 4c+1), lanes 16-31 (K=4c+2, 4c+3)
            const float s0 = __shfl(fr[4 * c + 0], src, 32);
            const float s1 = __shfl(fr[4 * c + 1], src, 32);
            const float s2v = __shfl(fr[4 * c + 2], src, 32);
            const float s3v = __shfl(fr[4 * c + 3], src, 32);
            v2f B;
            B.x = hi ? s2v : s0;
            B.y = hi ? s3v : s1;
            acc = __builtin_amdgcn_wmma_f32_16x16x4_f32(
                /*neg_a=*/false, A[c], /*neg_b=*/false, B,
                /*c_mod=*/(short)0, acc, /*reuse_a=*/false, /*reuse_b=*/false);
        }

        const long long q = base + (g << 4) + row;
        if (q < N) {
            const size_t qq = (size_t)q;
            if (!hi) {
                // rows 0..7: Y0, Y1[0..2], Y2[0..3]
                __builtin_nontemporal_store(acc[0], out  + qq);
                __builtin_nontemporal_store(acc[1], out1 + 3 * qq + 0);
                __builtin_nontemporal_store(acc[2], out1 + 3 * qq + 1);
                __builtin_nontemporal_store(acc[3], out1 + 3 * qq + 2);
                __builtin_nontemporal_store(acc[4], out2 + 5 * qq + 0);
                __builtin_nontemporal_store(acc[5], out2 + 5 * qq + 1);
                __builtin_nontemporal_store(acc[6], out2 + 5 * qq + 2);
                __builtin_nontemporal_store(acc[7], out2 + 5 * qq + 3);
            } else {
                // rows 8..15: Y2[4], Y3[0..6]
                __builtin_nontemporal_store(acc[0], out2 + 5 * qq + 4);
                __builtin_nontemporal_store(acc[1], out3 + 7 * qq + 0);
                __builtin_nontemporal_store(acc[2], out3 + 7 * qq + 1);
                __builtin_nontemporal_store(acc[3], out3 + 7 * qq + 2);
                __builtin_nontemporal_store(acc[4], out3 + 7 * qq + 3);
                __builtin_nontemporal_store(acc[5], out3 + 7 * qq + 4);
                __builtin_nontemporal_store(acc[6], out3 + 7 * qq + 5);
                __builtin_nontemporal_store(acc[7], out3 + 7 * qq + 6);
            }
        }
    }
}

extern "C" void kernel_launch(void* const* d_in, const int* in_sizes, int n_in,
                              void* d_out, int out_size, void* d_ws, size_t ws_size,
                              hipStream_t stream) {
    const float* d = (const float*)d_in[0];
    float* out = (float*)d_out;
    const int N = in_sizes[0] / 3;              // 4,000,000
    const int pointsPerBlock = 256;             // 8 waves x 32 points
    const int blocks = (N + pointsPerBlock - 1) / pointsPerBlock;
    sph_wmma_kernel<<<blocks, 256, 0, stream>>>(d, out, N);
}